// DPAR_23295902613912
// MI455X (gfx1250) — compile-verified
//
#include <hip/hip_runtime.h>
#include <math.h>

#define N       16384
#define NFEAT   500
#define HIDDEN  32
#define NCLASS  3
#define ALPHA   0.25f

#define ROWS_PER_BLOCK 16
#define PROP_THREADS   512
// SoA staging of the [N,4] vector (4 * N floats) + 16 float4 reduction slots
#define PROP_LDS (4 * N * (int)sizeof(float) + 16 * (int)sizeof(float4))

typedef __attribute__((ext_vector_type(2))) float v2f;
typedef __attribute__((ext_vector_type(8))) float v8f;

// ---------------------------------------------------------------------------
// Kernel 1: local4[m] = { (relu(input@W1) @ W2)[m, 0..2], 1.0 }
// One wave (32 lanes) per 16-row tile, V_WMMA_F32_16X16X4_F32 over K=500.
// ---------------------------------------------------------------------------
__global__ __launch_bounds__(256) void gcn_local_kernel(
    const float* __restrict__ input,   // [N, NFEAT]
    const float* __restrict__ W1,      // [NFEAT, HIDDEN]
    const float* __restrict__ W2,      // [HIDDEN, NCLASS]
    float4* __restrict__ local4)       // [N]
{
    __shared__ float htile[8][16][HIDDEN];

    const int lane  = threadIdx.x & 31;
    const int wave  = threadIdx.x >> 5;
    const int tile  = blockIdx.x * 8 + wave;     // 1024 tiles total
    const int row_base = tile * 16;

    // A-matrix (16x4 f32) lane mapping per ISA:
    // lanes 0-15: M=lane,   VGPR0/1 = K,K+1 ; lanes 16-31: M=lane-16, K+2,K+3
    const int mrow  = row_base + (lane & 15);
    const int khalf = (lane >> 4) * 2;           // 0 or 2
    const int n0    = lane & 15;

    v8f acc0 = {0.f,0.f,0.f,0.f,0.f,0.f,0.f,0.f};
    v8f acc1 = {0.f,0.f,0.f,0.f,0.f,0.f,0.f,0.f};

    const float* arow = input + (size_t)mrow * NFEAT;
    for (int k = 0; k < NFEAT; k += 4) {
        // aligned float2: row stride 2000 B (mult of 8), k+khalf even
        v2f a = *(const v2f*)(arow + k + khalf);
        const int kb = k + khalf;                // B (4x16): VGPR v = K kb+v
        v2f b0, b1;
        b0.x = W1[kb * HIDDEN + n0];
        b0.y = W1[(kb + 1) * HIDDEN + n0];
        b1.x = W1[kb * HIDDEN + 16 + n0];
        b1.y = W1[(kb + 1) * HIDDEN + 16 + n0];
        acc0 = __builtin_amdgcn_wmma_f32_16x16x4_f32(false, a, false, b0,
                                                     (short)0, acc0, false, false);
        acc1 = __builtin_amdgcn_wmma_f32_16x16x4_f32(false, a, false, b1,
                                                     (short)0, acc1, false, false);
    }

    // D layout: VGPR r -> lanes 0-15:(M=r,N=lane), lanes 16-31:(M=r+8,N=lane-16)
#pragma unroll
    for (int r = 0; r < 8; ++r) {
        const int m = r + (lane >> 4) * 8;
        htile[wave][m][n0]      = fmaxf(acc0[r], 0.0f);   // relu
        htile[wave][m][n0 + 16] = fmaxf(acc1[r], 0.0f);
    }
    __syncthreads();

    // tiny tail: [16x32] @ [32x3] per wave, lanes 0-15 each own one row
    if (lane < 16) {
        const int m = row_base + lane;
        float c0 = 0.f, c1 = 0.f, c2 = 0.f;
#pragma unroll
        for (int n = 0; n < HIDDEN; ++n) {
            const float h = htile[wave][lane][n];
            c0 += h * W2[n * NCLASS + 0];
            c1 += h * W2[n * NCLASS + 1];
            c2 += h * W2[n * NCLASS + 2];
        }
        local4[m] = make_float4(c0, c1, c2, 1.0f);  // .w=1 -> fused degree sum
    }
}

// ---------------------------------------------------------------------------
// Kernel 2: one APPNP propagation pass.  vec[N] (float4) staged SoA in LDS
// (256 KB -> fits the 320 KB CDNA5 WGP LDS); each block streams 16 adj rows.
// phase 0: vec = local4 (w=1): y.w = deg; out1 = {deg_inv*(adj@L)+a*L, deg_inv}
// phase 1: vec = out1 (w=deg_inv): finalize + log_softmax -> outf [N,3]
// ---------------------------------------------------------------------------
__global__ __launch_bounds__(PROP_THREADS) void prop_kernel(
    const float* __restrict__ adj,      // [N, N]
    const float4* __restrict__ vec,     // [N]
    const float4* __restrict__ local4,  // [N]
    float4* __restrict__ out1,          // [N]  (phase 0)
    float* __restrict__ outf,           // [N,3](phase 1)
    int phase)
{
    extern __shared__ float smem[];
    float* sVx = smem;
    float* sVy = smem + N;
    float* sVz = smem + 2 * N;
    float* sVw = smem + 3 * N;
    float4* spart = (float4*)(smem + 4 * N);

    const int tid = threadIdx.x;

    // stage vec into LDS (SoA -> conflict-free b128 reads later)
    for (int c = tid; c < N; c += PROP_THREADS) {
        const float4 v = vec[c];
        sVx[c] = v.x; sVy[c] = v.y; sVz[c] = v.z; sVw[c] = v.w;
    }
    __syncthreads();

    const float4* vx4 = (const float4*)sVx;
    const float4* vy4 = (const float4*)sVy;
    const float4* vz4 = (const float4*)sVz;
    const float4* vw4 = (const float4*)sVw;

    for (int i = 0; i < ROWS_PER_BLOCK; ++i) {
        const int row = blockIdx.x * ROWS_PER_BLOCK + i;
        const float4* arow = (const float4*)(adj + (size_t)row * N);

        float4 acc = make_float4(0.f, 0.f, 0.f, 0.f);
        for (int c4 = tid; c4 < N / 4; c4 += PROP_THREADS) {
            const float4 a = arow[c4];
            const float4 x = vx4[c4];
            const float4 y = vy4[c4];
            const float4 z = vz4[c4];
            const float4 w = vw4[c4];
            acc.x += a.x * x.x + a.y * x.y + a.z * x.z + a.w * x.w;
            acc.y += a.x * y.x + a.y * y.y + a.z * y.z + a.w * y.w;
            acc.z += a.x * z.x + a.y * z.y + a.z * z.z + a.w * z.w;
            acc.w += a.x * w.x + a.y * w.y + a.z * w.z + a.w * w.w;
        }

        // wave32 reduction
#pragma unroll
        for (int off = 16; off > 0; off >>= 1) {
            acc.x += __shfl_down(acc.x, off, 32);
            acc.y += __shfl_down(acc.y, off, 32);
            acc.z += __shfl_down(acc.z, off, 32);
            acc.w += __shfl_down(acc.w, off, 32);
        }
        if ((tid & 31) == 0) spart[tid >> 5] = acc;
        __syncthreads();

        if (tid == 0) {
            float4 s = make_float4(0.f, 0.f, 0.f, 0.f);
#pragma unroll
            for (int wv = 0; wv < PROP_THREADS / 32; ++wv) {
                s.x += spart[wv].x; s.y += spart[wv].y;
                s.z += spart[wv].z; s.w += spart[wv].w;
            }
            if (phase == 0) {
                const float deg = s.w;                       // adj row-sum (free)
                const float di  = (1.0f - ALPHA) / fmaxf(deg, 1e-12f);
                const float lx = sVx[row], ly = sVy[row], lz = sVz[row];
                out1[row] = make_float4(di * s.x + ALPHA * lx,
                                        di * s.y + ALPHA * ly,
                                        di * s.z + ALPHA * lz,
                                        di);                 // carry deg_inv in .w
            } else {
                const float di = sVw[row];                   // deg_inv from pass 1
                const float4 l = local4[row];
                const float t0 = di * s.x + ALPHA * l.x;
                const float t1 = di * s.y + ALPHA * l.y;
                const float t2 = di * s.z + ALPHA * l.z;
                const float m  = fmaxf(t0, fmaxf(t1, t2));
                const float lse = logf(expf(t0 - m) + expf(t1 - m) + expf(t2 - m));
                outf[row * 3 + 0] = t0 - m - lse;
                outf[row * 3 + 1] = t1 - m - lse;
                outf[row * 3 + 2] = t2 - m - lse;
            }
        }
        __syncthreads();
    }
}

extern "C" void kernel_launch(void* const* d_in, const int* in_sizes, int n_in,
                              void* d_out, int out_size, void* d_ws, size_t ws_size,
                              hipStream_t stream) {
    const float* input = (const float*)d_in[0];   // [N, NFEAT]
    const float* adj   = (const float*)d_in[1];   // [N, N]
    const float* W1    = (const float*)d_in[2];   // [NFEAT, HIDDEN]
    const float* W2    = (const float*)d_in[3];   // [HIDDEN, NCLASS]
    float* out = (float*)d_out;                   // [N, NCLASS]

    float4* local4  = (float4*)d_ws;                                   // 256 KB
    float4* logits1 = (float4*)((char*)d_ws + (size_t)N * sizeof(float4)); // 256 KB

    (void)hipFuncSetAttribute(reinterpret_cast<const void*>(prop_kernel),
                              hipFuncAttributeMaxDynamicSharedMemorySize, PROP_LDS);

    // 1024 wave-tiles, 8 waves per 256-thread block
    gcn_local_kernel<<<N / 16 / 8, 256, 0, stream>>>(input, W1, W2, local4);

    // pass 1: fused degree + first propagation
    prop_kernel<<<N / ROWS_PER_BLOCK, PROP_THREADS, PROP_LDS, stream>>>(
        adj, local4, local4, logits1, nullptr, 0);

    // pass 2: second propagation + log_softmax
    prop_kernel<<<N / ROWS_PER_BLOCK, PROP_THREADS, PROP_LDS, stream>>>(
        adj, logits1, local4, nullptr, out, 1);
}